// PaiNN_27994596835761
// MI455X (gfx1250) — compile-verified
//
#include <hip/hip_runtime.h>

// ---------------------------------------------------------------------------
// PaiNN message layer for MI455X (gfx1250, wave32, WMMA f32 16x16x4).
// Memory-bound (~1.5 GB moved, ~65us at 23.3 TB/s); 12.7 GFLOP of f32 GEMM is
// pushed onto the matrix pipe (V_WMMA_F32_16X16X4_F32) so it stays hidden.
// v2: replace IEEE divides (v_div_scale_f32 chains) with native v_rcp_f32 in
//     ScaledSiLU and the per-edge 1/dist terms.
// ---------------------------------------------------------------------------

#define DD      128
#define TD      384          // 3*D
#define NRBF    20
#define CUTF    5.0f
#define PI_F    3.14159265358979323846f
#define SSCALE  (1.0f / 0.6f)

typedef __attribute__((ext_vector_type(2))) float v2f;
typedef __attribute__((ext_vector_type(8))) float v8f;

__device__ __forceinline__ v8f wmma_f32(v2f a, v2f b, v8f c) {
    // (neg_a, A, neg_b, B, c_mod, C, reuse_a, reuse_b)
    return __builtin_amdgcn_wmma_f32_16x16x4_f32(false, a, false, b,
                                                 (short)0, c, false, false);
}

__device__ __forceinline__ float fast_rcp(float x) {
    return __builtin_amdgcn_rcpf(x);       // v_rcp_f32
}

__device__ __forceinline__ void atomAddF32(float* p, float v) {
    unsafeAtomicAdd(p, v);   // global_atomic_add_f32 (no CAS loop)
}

// ---------------------------------------------------------------------------
__global__ void zero_f32(float* __restrict__ p, long n) {
    long i = (long)blockIdx.x * blockDim.x + threadIdx.x;
    long stride = (long)gridDim.x * blockDim.x;
    for (; i < n; i += stride) p[i] = 0.0f;
}

// ---------------------------------------------------------------------------
// s = scaled_silu(X @ W1 + b1) @ W2 + b2        X:[N,128] -> s:[N,384]
// One block (8 wave32) per 16 rows.
// ---------------------------------------------------------------------------
__global__ void __launch_bounds__(256)
node_proj_kernel(const float* __restrict__ x,
                 const float* __restrict__ W1, const float* __restrict__ b1,
                 const float* __restrict__ W2, const float* __restrict__ b2,
                 float* __restrict__ s_out, int Nn) {
    __shared__ float Xs[16][DD + 1];   // +1 pad: kill 16-way bank conflicts
    __shared__ float Hs[16][DD + 1];

    const int tid  = threadIdx.x;
    const int wave = tid >> 5;
    const int lane = tid & 31;
    const int j    = lane & 15;            // column / row-in-tile index
    const int half = (lane < 16) ? 0 : 2;  // K sub-offset per ISA A/B layout
    const int rbase = blockIdx.x * 16;

    // stage X tile into LDS, coalesced (256 thr * 8 = 2048 floats)
#pragma unroll
    for (int i = 0; i < 8; ++i) {
        int idx = tid + 256 * i;
        int r = idx >> 7, c = idx & 127;
        float v = (rbase + r < Nn) ? x[(size_t)(rbase + r) * DD + c] : 0.0f;
        Xs[r][c] = v;
    }
    __syncthreads();

    // ---- stage 1: H = scaled_silu(X @ W1 + b1); wave w -> cols [16w,16w+16)
    {
        const int col = 16 * wave + j;
        v8f acc = {};
#pragma unroll 8
        for (int kk = 0; kk < 32; ++kk) {
            const int k0 = 4 * kk + half;
            v2f a; a.x = Xs[j][k0]; a.y = Xs[j][k0 + 1];
            v2f b; b.x = W1[(size_t)k0 * DD + col];
                   b.y = W1[(size_t)(k0 + 1) * DD + col];
            acc = wmma_f32(a, b, acc);
        }
        const float bias = b1[col];
#pragma unroll
        for (int r = 0; r < 8; ++r) {
            int row = (lane < 16) ? r : (r + 8);
            float h = acc[r] + bias;
            // scaled SiLU via native v_exp_f32 + v_rcp_f32 (no IEEE divide)
            float sig = fast_rcp(1.0f + __expf(-h));
            Hs[row][col] = h * sig * SSCALE;
        }
    }
    __syncthreads();

    // ---- stage 2: S = H @ W2 + b2; wave w -> tiles {w, w+8, w+16} of 24
    v8f acc2[3] = {};
#pragma unroll 8
    for (int kk = 0; kk < 32; ++kk) {
        const int k0 = 4 * kk + half;
        v2f a; a.x = Hs[j][k0]; a.y = Hs[j][k0 + 1];
#pragma unroll
        for (int t = 0; t < 3; ++t) {
            const int col2 = 16 * (wave + 8 * t) + j;
            v2f b; b.x = W2[(size_t)k0 * TD + col2];
                   b.y = W2[(size_t)(k0 + 1) * TD + col2];
            acc2[t] = wmma_f32(a, b, acc2[t]);
        }
    }
#pragma unroll
    for (int t = 0; t < 3; ++t) {
        const int col2 = 16 * (wave + 8 * t) + j;
        const float bias2 = b2[col2];
#pragma unroll
        for (int r = 0; r < 8; ++r) {
            int row = (lane < 16) ? r : (r + 8);
            if (rbase + row < Nn)
                s_out[(size_t)(rbase + row) * TD + col2] = acc2[t][r] + bias2;
        }
    }
}

// ---------------------------------------------------------------------------
// Per 16-edge block: geometry + RBF, filt = cos_cutoff(rbf@Wr+br) via WMMA,
// gather s[src]/vf[src], fused scatter-add.  Wave w owns feature block
// d in [16w,16w+16) with channel-strided columns (d*3+ch), so each lane holds
// all three channels for one d -> one atomic per (e,d,xyz).
// ---------------------------------------------------------------------------
__global__ void __launch_bounds__(256)
edge_kernel(const float* __restrict__ vf,   const float* __restrict__ pos,
            const int*   __restrict__ src,  const int*   __restrict__ tgt,
            const float* __restrict__ s,    const float* __restrict__ Wr,
            const float* __restrict__ br,
            float* __restrict__ out_vec, float* __restrict__ out_scl, int Ee) {
    __shared__ float rbf_s[16][NRBF];
    __shared__ float dir_s[16][3];
    __shared__ float dist_s[16];
    __shared__ float inv_s[16];
    __shared__ int   se_s[16];
    __shared__ int   te_s[16];

    const int tid  = threadIdx.x;
    const int wave = tid >> 5;
    const int lane = tid & 31;
    const int j    = lane & 15;
    const int half = (lane < 16) ? 0 : 2;
    const int d    = 16 * wave + j;        // feature index owned by this lane
    const int ebase = blockIdx.x * 16;

    // preload B tiles (Wr, channel-strided columns) into registers + biases
    v2f Breg[3][5];
    float bias[3];
#pragma unroll
    for (int ch = 0; ch < 3; ++ch) {
        const int col = d * 3 + ch;
        bias[ch] = br[col];
#pragma unroll
        for (int kk = 0; kk < 5; ++kk) {
            const int k0 = 4 * kk + half;          // max 18, k0+1 <= 19 < 20
            Breg[ch][kk].x = Wr[(size_t)k0 * TD + col];
            Breg[ch][kk].y = Wr[(size_t)(k0 + 1) * TD + col];
        }
    }

    // ---- geometry (16 lanes)
    if (tid < 16) {
        const int e = ebase + tid;
        if (e < Ee) {
            const int a = src[e], b = tgt[e];
            se_s[tid] = a; te_s[tid] = b;
            float rx = pos[3 * b + 0] - pos[3 * a + 0];
            float ry = pos[3 * b + 1] - pos[3 * a + 1];
            float rz = pos[3 * b + 2] - pos[3 * a + 2];
            float dist = sqrtf(rx * rx + ry * ry + rz * rz);
            float inv = fast_rcp(dist);            // dist > 0 (no self loops)
            dist_s[tid] = dist;
            inv_s[tid]  = inv;
            dir_s[tid][0] = rx * inv; dir_s[tid][1] = ry * inv; dir_s[tid][2] = rz * inv;
        } else {
            se_s[tid] = 0; te_s[tid] = -1;  // invalid -> skipped in scatter
            dist_s[tid] = 1.0f; inv_s[tid] = 1.0f;
            dir_s[tid][0] = dir_s[tid][1] = dir_s[tid][2] = 0.0f;
        }
    }
    __syncthreads();

    // ---- rbf tile: 16 edges x 20 features spread over 256 threads
    for (int idx = tid; idx < 16 * NRBF; idx += 256) {
        const int e = idx / NRBF, k = idx % NRBF;
        rbf_s[e][k] = __sinf((float)(k + 1) * PI_F * dist_s[e] * (1.0f / CUTF))
                      * inv_s[e];
    }
    __syncthreads();

    // ---- filt = rbf @ Wr  (K = 20 = 5 WMMA k-steps), 3 channel tiles/wave
    v8f acc[3] = {};
#pragma unroll
    for (int kk = 0; kk < 5; ++kk) {
        const int k0 = 4 * kk + half;
        v2f a; a.x = rbf_s[j][k0]; a.y = rbf_s[j][k0 + 1];
#pragma unroll
        for (int ch = 0; ch < 3; ++ch)
            acc[ch] = wmma_f32(a, Breg[ch][kk], acc[ch]);
    }

    // ---- bias + cosine cutoff, gather, fused scatter-add
#pragma unroll
    for (int r = 0; r < 8; ++r) {
        const int row = (lane < 16) ? r : (r + 8);
        const int te = te_s[row];
        if (te < 0) continue;
        const int se = se_s[row];

        float m[3];
        const float* sp = s + (size_t)se * TD + 3 * d;
#pragma unroll
        for (int ch = 0; ch < 3; ++ch) {
            float y = acc[ch][r] + bias[ch];
            float f = (y < CUTF) ? 0.5f * (1.0f + __cosf(y * (PI_F / CUTF))) : 0.0f;
            m[ch] = sp[ch] * f;               // message = s[src] * filt
        }

        // scalar channel
        atomAddF32(&out_scl[(size_t)te * DD + d], m[0]);

        // vector channels fused: invar*dir + equiv*vf[src]
        const float* vp = vf + ((size_t)se * DD + d) * 3;
        const float vx = vp[0], vy = vp[1], vz = vp[2];
        const float dx = dir_s[row][0], dy = dir_s[row][1], dz = dir_s[row][2];
        float* op = out_vec + ((size_t)te * DD + d) * 3;
        atomAddF32(&op[0], m[2] * dx + m[1] * vx);
        atomAddF32(&op[1], m[2] * dy + m[1] * vy);
        atomAddF32(&op[2], m[2] * dz + m[1] * vz);
    }
}

// ---------------------------------------------------------------------------
extern "C" void kernel_launch(void* const* d_in, const int* in_sizes, int n_in,
                              void* d_out, int out_size, void* d_ws, size_t ws_size,
                              hipStream_t stream) {
    (void)n_in; (void)ws_size;
    const float* vf  = (const float*)d_in[0];   // [N,D,3]
    const float* sf  = (const float*)d_in[1];   // [N,D]
    const float* pos = (const float*)d_in[2];   // [N,3]
    const int*   ei  = (const int*)d_in[3];     // [2,E]
    const float* W1  = (const float*)d_in[4];
    const float* b1  = (const float*)d_in[5];
    const float* W2  = (const float*)d_in[6];
    const float* b2  = (const float*)d_in[7];
    const float* Wr  = (const float*)d_in[8];
    const float* br  = (const float*)d_in[9];

    const int Nn = in_sizes[1] / DD;
    const int Ee = in_sizes[3] / 2;

    float* out_vec = (float*)d_out;                      // [N,D,3]
    float* out_scl = out_vec + (size_t)Nn * DD * 3;      // [N,D]
    float* s_ws    = (float*)d_ws;                       // [N,384] scratch

    // zero outputs (atomics accumulate into them)
    {
        long n = (long)out_size;
        int blocks = (int)((n + 256L * 16 - 1) / (256L * 16));
        if (blocks > 16384) blocks = 16384;
        if (blocks < 1) blocks = 1;
        zero_f32<<<blocks, 256, 0, stream>>>((float*)d_out, n);
    }

    // node projection s = scaled_silu(X@W1+b1)@W2+b2
    node_proj_kernel<<<(Nn + 15) / 16, 256, 0, stream>>>(sf, W1, b1, W2, b2,
                                                         s_ws, Nn);

    // per-edge filter + gather + fused scatter
    edge_kernel<<<(Ee + 15) / 16, 256, 0, stream>>>(vf, pos, ei, ei + Ee,
                                                    s_ws, Wr, br,
                                                    out_vec, out_scl, Ee);
}